// ContrastiveLoss_cosine2_61340722921803
// MI455X (gfx1250) — compile-verified
//
#include <hip/hip_runtime.h>

typedef __attribute__((ext_vector_type(2))) float v2f;
typedef __attribute__((ext_vector_type(8))) float v8f;

#define EPS_     1e-8f
#define MARGIN_  0.2f

// ---------------------------------------------------------------------------
// Kernel 1: one wave per 16-row tile. Computes diag(Img·TxtT), diag(Img·ImgT),
// diag(Txt·TxtT) via V_WMMA_F32_16X16X4_F32 chains (K permuted so each lane
// owns a contiguous half-row -> pure float4 streaming loads).
// Emits: norms, pos_dist, and both cosine output vectors.
// ---------------------------------------------------------------------------
__global__ __launch_bounds__(256) void k_rowdots_wmma(
    const float* __restrict__ img, const float* __restrict__ txt,
    float* __restrict__ nimg, float* __restrict__ ntxt,
    float* __restrict__ posd,
    float* __restrict__ cos_i2t, float* __restrict__ cos_t2i,
    int B, int D) {
  const int lane = threadIdx.x & 31;
  const int wave = threadIdx.x >> 5;
  const int tile = blockIdx.x * (blockDim.x >> 5) + wave;
  if (tile * 16 >= B) return;                 // wave-uniform guard (EXEC stays full)

  const int m = lane & 15;                    // A row index == B col index for this lane
  const int h = lane >> 4;                    // which half-row this lane owns
  const int halfD = D >> 1;
  const float4* xrow = (const float4*)(img + (size_t)(tile * 16 + m) * D + h * halfD);
  const float4* yrow = (const float4*)(txt + (size_t)(tile * 16 + m) * D + h * halfD);

  v8f c_it = {}; v8f c_ii = {}; v8f c_tt = {};
  const int steps = halfD >> 2;               // float4s per half-row (128)
  for (int s = 0; s < steps; ++s) {
    float4 xa = xrow[s];
    float4 ya = yrow[s];
    v2f a0 = {xa.x, xa.y}, a1 = {xa.z, xa.w};
    v2f b0 = {ya.x, ya.y}, b1 = {ya.z, ya.w};
    c_it = __builtin_amdgcn_wmma_f32_16x16x4_f32(false, a0, false, b0, (short)0, c_it, false, false);
    c_ii = __builtin_amdgcn_wmma_f32_16x16x4_f32(false, a0, false, a0, (short)0, c_ii, false, false);
    c_tt = __builtin_amdgcn_wmma_f32_16x16x4_f32(false, b0, false, b0, (short)0, c_tt, false, false);
    c_it = __builtin_amdgcn_wmma_f32_16x16x4_f32(false, a1, false, b1, (short)0, c_it, false, false);
    c_ii = __builtin_amdgcn_wmma_f32_16x16x4_f32(false, a1, false, a1, (short)0, c_ii, false, false);
    c_tt = __builtin_amdgcn_wmma_f32_16x16x4_f32(false, b1, false, b1, (short)0, c_tt, false, false);
  }

  // Diagonal of C (16x16 f32): row r<8 -> lane r, comp r; row r>=8 -> lane r+16, comp r-8.
  int rowoff = -1, comp = 0;
  if (lane < 8)        { rowoff = lane;      comp = lane;      }
  else if (lane >= 24) { rowoff = lane - 16; comp = lane - 24; }
  if (rowoff >= 0) {
    float wit = 0.f, wii = 0.f, wtt = 0.f;
#pragma unroll
    for (int i = 0; i < 8; ++i) {
      wit = (comp == i) ? c_it[i] : wit;
      wii = (comp == i) ? c_ii[i] : wii;
      wtt = (comp == i) ? c_tt[i] : wtt;
    }
    const int row  = tile * 16 + rowoff;
    const float ni = sqrtf(wii);
    const float nt = sqrtf(wtt);
    const float dn = ni * nt;
    const float cosv = wit / dn;                       // unclamped, like reference sim
    const float pd   = 1.0f - wit / fmaxf(dn, EPS_);   // clamped, like reference dist
    nimg[row] = ni;
    ntxt[row] = nt;
    posd[row] = pd;
    cos_i2t[row] = cosv;
    cos_t2i[row] = cosv;                               // identical by symmetry
  }
}

// ---------------------------------------------------------------------------
// Kernel 2: one wave per row. 4 gathered candidate dots (float4 loads, wave32
// shuffle reduce), then hard-negative select (<= : later candidate wins ties)
// and per-row triplet terms.
// ---------------------------------------------------------------------------
__device__ __forceinline__ float dot4_(float4 a, float4 b) {
  return a.x * b.x + a.y * b.y + a.z * b.z + a.w * b.w;
}

__global__ __launch_bounds__(256) void k_neg_terms(
    const float* __restrict__ img, const float* __restrict__ txt,
    const int* __restrict__ cand_img, const int* __restrict__ cand_txt,
    const float* __restrict__ nimg, const float* __restrict__ ntxt,
    const float* __restrict__ posd,
    float* __restrict__ termI2T, float* __restrict__ termT2I,
    int B, int D) {
  const int lane = threadIdx.x & 31;
  const int wave = threadIdx.x >> 5;
  const int row  = blockIdx.x * (blockDim.x >> 5) + wave;
  if (row >= B) return;

  const int c0 = cand_img[2 * row], c1 = cand_img[2 * row + 1];
  const int d0 = cand_txt[2 * row], d1 = cand_txt[2 * row + 1];

  const float4* tb = (const float4*)(txt + (size_t)row * D);
  const float4* ib = (const float4*)(img + (size_t)row * D);
  const float4* i0 = (const float4*)(img + (size_t)c0 * D);
  const float4* i1 = (const float4*)(img + (size_t)c1 * D);
  const float4* t0 = (const float4*)(txt + (size_t)d0 * D);
  const float4* t1 = (const float4*)(txt + (size_t)d1 * D);

  const int nv = D >> 2;
  float ai0 = 0.f, ai1 = 0.f, at0 = 0.f, at1 = 0.f;
  for (int j = lane; j < nv; j += 32) {
    float4 a = tb[j], b = ib[j];
    ai0 += dot4_(a, i0[j]);
    ai1 += dot4_(a, i1[j]);
    at0 += dot4_(b, t0[j]);
    at1 += dot4_(b, t1[j]);
  }
#pragma unroll
  for (int off = 16; off > 0; off >>= 1) {
    ai0 += __shfl_down(ai0, off, 32);
    ai1 += __shfl_down(ai1, off, 32);
    at0 += __shfl_down(at0, off, 32);
    at1 += __shfl_down(at1, off, 32);
  }
  if (lane == 0) {
    const float nt = ntxt[row], ni = nimg[row];
    // image-side negatives: anchor = text[row], embedding = img
    const float di0 = 1.f - ai0 / fmaxf(nt * nimg[c0], EPS_);
    const float di1 = 1.f - ai1 / fmaxf(nt * nimg[c1], EPS_);
    const float t2i_neg = (di1 <= di0) ? di1 : di0;
    // text-side negatives: anchor = img[row], embedding = txt
    const float dt0 = 1.f - at0 / fmaxf(ni * ntxt[d0], EPS_);
    const float dt1 = 1.f - at1 / fmaxf(ni * ntxt[d1], EPS_);
    const float i2t_neg = (dt1 <= dt0) ? dt1 : dt0;

    const float pd = posd[row];                 // pos_dist == t2i_pos_dist (symmetric)
    termI2T[row] = fmaxf(pd - i2t_neg + MARGIN_, 0.f);
    termT2I[row] = fmaxf(pd - t2i_neg + MARGIN_, 0.f);
  }
}

// ---------------------------------------------------------------------------
// Kernel 3: deterministic fixed-order reduction -> scalar loss.
// ---------------------------------------------------------------------------
__global__ __launch_bounds__(256) void k_loss(
    const float* __restrict__ termI2T, const float* __restrict__ termT2I,
    float* __restrict__ out, int B) {
  __shared__ float s1[256];
  __shared__ float s2[256];
  const int t = threadIdx.x;
  float a = 0.f, b = 0.f;
  for (int j = t; j < B; j += 256) { a += termI2T[j]; b += termT2I[j]; }
  s1[t] = a; s2[t] = b;
  __syncthreads();
#pragma unroll
  for (int off = 128; off > 0; off >>= 1) {
    if (t < off) { s1[t] += s1[t + off]; s2[t] += s2[t + off]; }
    __syncthreads();
  }
  if (t == 0) out[0] = (s1[0] + s2[0]) / (float)B;
}

// ---------------------------------------------------------------------------
// Launch
// ---------------------------------------------------------------------------
extern "C" void kernel_launch(void* const* d_in, const int* in_sizes, int n_in,
                              void* d_out, int out_size, void* d_ws, size_t ws_size,
                              hipStream_t stream) {
  const float* img      = (const float*)d_in[0];   // [B,D]
  const float* txt      = (const float*)d_in[1];   // [B,D]
  // d_in[2] = labels (unused), d_in[3] = locations (unused)
  const int*   cand_img = (const int*)d_in[4];     // [B,2]
  const int*   cand_txt = (const int*)d_in[5];     // [B,2]

  const int B = in_sizes[2];                       // labels: [B]
  const int D = in_sizes[0] / B;                   // 1024

  float* out      = (float*)d_out;                 // [0]=loss, [1..B]=i2t_cos, [1+B..2B]=t2i_cos
  float* cos_i2t  = out + 1;
  float* cos_t2i  = out + 1 + B;

  float* w      = (float*)d_ws;                    // 5*B floats = 80 KB
  float* nimg   = w;
  float* ntxt   = w + B;
  float* posd   = w + 2 * (size_t)B;
  float* termA  = w + 3 * (size_t)B;               // i2t terms
  float* termB  = w + 4 * (size_t)B;               // t2i terms

  const int wavesPerBlock = 8;                     // 256 threads = 8 wave32
  const int tiles  = (B + 15) / 16;
  const int grid1  = (tiles + wavesPerBlock - 1) / wavesPerBlock;
  const int grid2  = (B + wavesPerBlock - 1) / wavesPerBlock;

  k_rowdots_wmma<<<grid1, 256, 0, stream>>>(img, txt, nimg, ntxt, posd,
                                            cos_i2t, cos_t2i, B, D);
  k_neg_terms<<<grid2, 256, 0, stream>>>(img, txt, cand_img, cand_txt,
                                         nimg, ntxt, posd, termA, termB, B, D);
  k_loss<<<1, 256, 0, stream>>>(termA, termB, out, B);
}